// TraitSimilarity_45518063403661
// MI455X (gfx1250) — compile-verified
//
#include <hip/hip_runtime.h>

typedef float v2f __attribute__((ext_vector_type(2)));
typedef float v8f __attribute__((ext_vector_type(8)));

// D = A(16x4) * B(4x16) + C, fp32, wave32
#define WMMA4(A, B, C) \
  __builtin_amdgcn_wmma_f32_16x16x4_f32(false, (A), false, (B), (short)0, (C), false, false)

#define GRAM_BLOCKS  512
#define GRAM_THREADS 256
// ws layout (floats): [0,1024) G_true, [1024,2048) G_pred, [2048,2080) col sums of y_true
#define WS_FLOATS 2080

__global__ void ts_zero_ws(float* ws) {
  int i = blockIdx.x * blockDim.x + threadIdx.x;
  if (i < WS_FLOATS) ws[i] = 0.0f;
}

__global__ __launch_bounds__(GRAM_THREADS)
void ts_gram(const float* __restrict__ pred, const float* __restrict__ tru,
             float* __restrict__ ws, int nSamples) {
  __shared__ float lds[WS_FLOATS];
  for (int i = threadIdx.x; i < WS_FLOATS; i += blockDim.x) lds[i] = 0.0f;
  __syncthreads();

  const int lane = threadIdx.x & 31;
  const int col  = lane & 15;   // column within a 16-wide block
  const int half = lane >> 4;   // 0: samples n0,n1   1: samples n2,n3
  const int wavesPerBlock = GRAM_THREADS >> 5;
  const int gw     = blockIdx.x * wavesPerBlock + (threadIdx.x >> 5);
  const int stride = gridDim.x * wavesPerBlock * 4;

  // 2x2 blocks of the 32x32 Gram, per matrix; 8 VGPRs each
  v8f t00 = {}, t01 = {}, t10 = {}, t11 = {};
  v8f p00 = {}, p01 = {}, p10 = {}, p11 = {};
  float s0 = 0.0f, s1 = 0.0f;  // partial column sums of y_true

  // Wave-uniform loop: EXEC is all-1s around every WMMA.
  for (int n = gw * 4; n < nSamples; n += stride) {
    const long b0 = (long)(n + 2 * half) * 32;  // row for VGPR0 of fragments
    const long b1 = b0 + 32;                    // row for VGPR1
    v2f tF0, tF1, pF0, pF1;
    tF0.x = tru[b0 + col];       tF0.y = tru[b1 + col];
    tF1.x = tru[b0 + col + 16];  tF1.y = tru[b1 + col + 16];
    pF0.x = pred[b0 + col];      pF0.y = pred[b1 + col];
    pF1.x = pred[b0 + col + 16]; pF1.y = pred[b1 + col + 16];

    s0 += tF0.x + tF0.y;
    s1 += tF1.x + tF1.y;

    t00 = WMMA4(tF0, tF0, t00);
    t01 = WMMA4(tF0, tF1, t01);
    t10 = WMMA4(tF1, tF0, t10);
    t11 = WMMA4(tF1, tF1, t11);
    p00 = WMMA4(pF0, pF0, p00);
    p01 = WMMA4(pF0, pF1, p01);
    p10 = WMMA4(pF1, pF0, p10);
    p11 = WMMA4(pF1, pF1, p11);
  }

  // Reduce this wave's partials into block-shared LDS (ds_add_f32).
  atomicAdd(&lds[2048 + col],      s0);  // both halves contribute (diff samples)
  atomicAdd(&lds[2048 + 16 + col], s1);
#pragma unroll
  for (int r = 0; r < 8; ++r) {
    // C/D layout: VGPR r, lanes 0-15 -> M=r, N=lane; lanes 16-31 -> M=r+8
    const int m = r + 8 * half;
    atomicAdd(&lds[(m)      * 32 + col],       t00[r]);
    atomicAdd(&lds[(m)      * 32 + col + 16],  t01[r]);
    atomicAdd(&lds[(m + 16) * 32 + col],       t10[r]);
    atomicAdd(&lds[(m + 16) * 32 + col + 16],  t11[r]);
    atomicAdd(&lds[1024 + (m)      * 32 + col],      p00[r]);
    atomicAdd(&lds[1024 + (m)      * 32 + col + 16], p01[r]);
    atomicAdd(&lds[1024 + (m + 16) * 32 + col],      p10[r]);
    atomicAdd(&lds[1024 + (m + 16) * 32 + col + 16], p11[r]);
  }
  __syncthreads();

  // One global atomic per element per block (global_atomic_add_f32).
  for (int i = threadIdx.x; i < WS_FLOATS; i += blockDim.x)
    atomicAdd(&ws[i], lds[i]);
}

__global__ __launch_bounds__(1024)
void ts_finalize(const float* __restrict__ ws, float* __restrict__ out, int nSamples) {
  const float* Gt = ws;
  const float* Gp = ws + 1024;
  const float* S  = ws + 2048;
  const int j = threadIdx.x >> 5;
  const int k = threadIdx.x & 31;
  const float invN = 1.0f / (float)nSamples;
  float v = 0.0f;
  if (j >= 1 && k > j) {
    // centered Gram via G - S_j*S_k/N (one-pass mean removal)
    const float gjj = Gt[j * 32 + j] - S[j] * S[j] * invN;
    const float gkk = Gt[k * 32 + k] - S[k] * S[k] * invN;
    const float gjk = Gt[j * 32 + k] - S[j] * S[k] * invN;
    const float pcc = gjk / (sqrtf(gjj) * sqrtf(gkk));
    if (pcc >= 0.0f) {  // DELTA = 0.0
      const float denom = fmaxf(sqrtf(Gp[j * 32 + j]) * sqrtf(Gp[k * 32 + k]), 1e-8f);
      v = 1.0f - Gp[j * 32 + k] / denom;
    }
  }
  __shared__ float red[1024];
  red[threadIdx.x] = v;
  __syncthreads();
  for (int s = 512; s > 0; s >>= 1) {
    if (threadIdx.x < s) red[threadIdx.x] += red[threadIdx.x + s];
    __syncthreads();
  }
  if (threadIdx.x == 0) out[0] = red[0] * (1.0f / 465.0f);  // c = 31*30/2
}

extern "C" void kernel_launch(void* const* d_in, const int* in_sizes, int n_in,
                              void* d_out, int out_size, void* d_ws, size_t ws_size,
                              hipStream_t stream) {
  const float* y_pred = (const float*)d_in[0];
  const float* y_true = (const float*)d_in[1];
  float* ws  = (float*)d_ws;
  float* out = (float*)d_out;
  const int nSamples = in_sizes[0] / 32;  // T = 32

  ts_zero_ws<<<(WS_FLOATS + 255) / 256, 256, 0, stream>>>(ws);
  ts_gram<<<GRAM_BLOCKS, GRAM_THREADS, 0, stream>>>(y_pred, y_true, ws, nSamples);
  ts_finalize<<<1, 1024, 0, stream>>>(ws, out, nSamples);
}